// EGTLayer_53343493816853
// MI455X (gfx1250) — compile-verified
//
#include <hip/hip_runtime.h>
#include <hip/hip_bf16.h>
#include <math.h>
#include <stdint.h>

// EGT layer, MI455X (gfx1250, wave32). GEMMs via V_WMMA_F32_16X16X4_F32
// (memory-bound -> stay fp32, engage matrix pipes at no precision cost).
// Bulk efeat streaming via Tensor Data Mover (tensor_load_to_lds) with LDS
// padding (stride 36 dwords) for bank-conflict-free per-entry reads.

#define Bx 8
#define Nn 512
#define Ff 128
#define Ee 32
#define Hh 8
#define Dd 16
#define NVi 4

typedef float v2f __attribute__((ext_vector_type(2)));
typedef float v8f __attribute__((ext_vector_type(8)));
typedef unsigned int u32x4 __attribute__((ext_vector_type(4)));
typedef int i32x8 __attribute__((ext_vector_type(8)));
typedef int i32x4 __attribute__((ext_vector_type(4)));

// workspace layout (float offsets)
constexpr size_t Q_OFF    = 0;          // [B][H][N][D] = 524288
constexpr size_t K_OFF    = 524288;
constexpr size_t V_OFF    = 1048576;
constexpr size_t ATTN_OFF = 1572864;    // [B][H][N][N] = 16777216
constexpr size_t GATE_OFF = 18350080;   // [B][H][N][N] = 16777216
constexpr size_t RMAX_OFF = 35127296;   // [B][H][N] = 32768
constexpr size_t RSUM_OFF = 35160064;
constexpr size_t SCAL_OFF = 35192832;
constexpr size_t VATT_OFF = 35225600;   // [B][N][F] = 524288
constexpr size_t NODE_SZ  = 524288;     // n_out floats in d_out; e_out follows

#define PADW 36  // LDS row stride in dwords: 32 data + 4 TDM pad

__device__ __forceinline__ v8f wmma4(v2f a, v2f b, v8f c) {
  return __builtin_amdgcn_wmma_f32_16x16x4_f32(false, a, false, b, (short)0, c,
                                               false, false);
}

// Tensor DMA: 2-D tile (tile_d0 x tile_d1 elements, 4B/elem) from global to
// LDS, row stride stride0 elements, with padding 4 dwords per 32 dwords.
__device__ __forceinline__ void tdm_load_2d(const void* gaddr, void* lds,
                                            unsigned tensor_d0, unsigned tensor_d1,
                                            unsigned tile_d0, unsigned tile_d1,
                                            unsigned long long stride0)
{
  unsigned long long ga = (unsigned long long)(uintptr_t)gaddr;
  unsigned la = (unsigned)(uintptr_t)lds;
  u32x4 g0;
  g0[0] = 1u;                                   // count=1, user descriptor
  g0[1] = la;                                   // lds_addr
  g0[2] = (unsigned)ga;                         // global_addr[31:0]
  g0[3] = (unsigned)((ga >> 32) & 0x01FFFFFFu)  // global_addr[56:32]
          | (2u << 30);                         // type=2 (image)
  i32x8 g1;
  g1[0] = (int)((2u << 16)      // data_size = 4B
                | (1u << 20)    // pad_enable
                | (4u << 22)    // pad_interval: 32 dwords
                | (3u << 25));  // pad_amount: 4 dwords
  g1[1] = (int)((tensor_d0 & 0xFFFFu) << 16);                     // dim0[15:0]
  g1[2] = (int)((tensor_d0 >> 16) | ((tensor_d1 & 0xFFFFu) << 16));
  g1[3] = (int)((tensor_d1 >> 16) | ((tile_d0 & 0xFFFFu) << 16));
  g1[4] = (int)(tile_d1 & 0xFFFFu);                               // tile_dim1
  g1[5] = (int)(stride0 & 0xFFFFFFFFull);                         // dim0 stride
  g1[6] = (int)((stride0 >> 32) & 0xFFFFull);
  g1[7] = 0;
  i32x4 z4 = {0, 0, 0, 0};
  i32x8 z8 = {0, 0, 0, 0, 0, 0, 0, 0};
  // 6-arg form (clang-23 / therock headers): groups 2/3 + extra group + cpol
  __builtin_amdgcn_tensor_load_to_lds(g0, g1, z4, z4, z8, 0);
}

// ---------------- kernel 1: node LN + QKV GEMM ----------------
__global__ __launch_bounds__(256) void k_node_pre(
    const float* __restrict__ nfeat, const float* __restrict__ g,
    const float* __restrict__ bb, const float* __restrict__ Wqkv,
    const float* __restrict__ bqkv, float* __restrict__ ws)
{
  __shared__ float sh[16][Ff + 1];
  __shared__ float red1[16][16], red2[16][16];
  __shared__ float smean[16], srstd[16];
  int blk = blockIdx.x;
  int b = blk >> 5;
  int r0 = (blk & 31) << 4;
  int t = threadIdx.x;
  int row = t >> 4, sub = t & 15;
  float vals[8];
  float s = 0.f, s2 = 0.f;
  const float* src = nfeat + ((size_t)(b * Nn + r0 + row)) * Ff + sub * 8;
#pragma unroll
  for (int i = 0; i < 8; i++) { float x = src[i]; vals[i] = x; s += x; s2 += x * x; }
  red1[row][sub] = s; red2[row][sub] = s2;
  __syncthreads();
  if (sub == 0) {
    float a = 0.f, c2 = 0.f;
#pragma unroll
    for (int i = 0; i < 16; i++) { a += red1[row][i]; c2 += red2[row][i]; }
    float m = a / 128.f, v = c2 / 128.f - m * m;
    smean[row] = m; srstd[row] = rsqrtf(v + 1e-5f);
  }
  __syncthreads();
  float m = smean[row], r = srstd[row];
#pragma unroll
  for (int i = 0; i < 8; i++) {
    int col = sub * 8 + i;
    sh[row][col] = (vals[i] - m) * r * g[col] + bb[col];
  }
  __syncthreads();
  int wave = t >> 5, lane = t & 31, half = lane >> 4, l16 = lane & 15;
  for (int ct = wave * 3; ct < wave * 3 + 3; ++ct) {
    int col0 = ct << 4;
    v8f c = {};
#pragma unroll
    for (int kk = 0; kk < Ff; kk += 4) {
      int ka = kk + 2 * half;
      v2f a, bv;
      a.x = sh[l16][ka]; a.y = sh[l16][ka + 1];
      bv.x = Wqkv[ka * 384 + col0 + l16];
      bv.y = Wqkv[(ka + 1) * 384 + col0 + l16];
      c = wmma4(a, bv, c);
    }
    int colg = col0 + l16;
    int sec = colg >> 7;        // 0=q 1=k 2=v
    int f = colg & 127;
    int d = f >> 3, hh = f & 7;
    float bias = bqkv[colg];
    float* dst = ws + (size_t)sec * 524288;
#pragma unroll
    for (int j = 0; j < 8; j++) {
      int rr = r0 + j + 8 * half;
      dst[(((size_t)b * Hh + hh) * Nn + rr) * Dd + d] = c[j] + bias;
    }
  }
}

// ---------------- kernel 2: edge LN + bias/gates + QK^T scores ----------------
__global__ __launch_bounds__(256) void k_edge_scores(
    const float* __restrict__ efeat, const float* __restrict__ eg,
    const float* __restrict__ ebv, const float* __restrict__ Wedge,
    const float* __restrict__ bedge, const float* __restrict__ Wgate,
    const float* __restrict__ bgate, float* __restrict__ ws)
{
  __shared__ float sTile[256 * PADW];       // 16x16 entries x 32 feats, TDM-padded
  __shared__ float s_ebias[16][16][Hh];
  int b = blockIdx.x, lt = blockIdx.y, mt = blockIdx.z;
  int l0 = lt << 4, m0 = mt << 4;
  int t = threadIdx.x;
  int ll = t >> 4, mm = t & 15;

  // Stream the 32KB edge tile via the Tensor Data Mover (one op per block):
  // 16 rows (l) of 512 contiguous floats (16 m-entries x 32), row stride N*E.
  const float* gsrc = efeat + ((((size_t)b * Nn + l0) * Nn) + m0) * Ee;
  if (t == 0) {
    tdm_load_2d(gsrc, sTile, Nn * Ee, Nn, 16 * Ee, 16,
                (unsigned long long)Nn * Ee);
  }
  __builtin_amdgcn_s_wait_tensorcnt(0);
  __syncthreads();

  const float4* ev = (const float4*)&sTile[(ll * 16 + mm) * PADW];
  float e[32];
  float s = 0.f, s2 = 0.f;
#pragma unroll
  for (int i = 0; i < 8; i++) {
    float4 q4 = ev[i];
    e[4 * i + 0] = q4.x; e[4 * i + 1] = q4.y;
    e[4 * i + 2] = q4.z; e[4 * i + 3] = q4.w;
    s += q4.x + q4.y + q4.z + q4.w;
    s2 += q4.x * q4.x + q4.y * q4.y + q4.z * q4.z + q4.w * q4.w;
  }
  float mean = s / 32.f, var = s2 / 32.f - mean * mean, rs = rsqrtf(var + 1e-5f);
#pragma unroll
  for (int i = 0; i < 32; i++) e[i] = (e[i] - mean) * rs * eg[i] + ebv[i];
  float* gates = ws + GATE_OFF;
#pragma unroll
  for (int h = 0; h < 8; h++) {
    float acc = bedge[h], acg = bgate[h];
#pragma unroll
    for (int i = 0; i < 32; i++) { acc += e[i] * Wedge[i * 8 + h]; acg += e[i] * Wgate[i * 8 + h]; }
    s_ebias[ll][mm][h] = acc;
    gates[((((size_t)b * Hh + h) * Nn) + l0 + ll) * Nn + m0 + mm] =
        1.f / (1.f + __expf(-acg));
  }
  __syncthreads();
  int wave = t >> 5, lane = t & 31, half = lane >> 4, l16 = lane & 15;
  int hh = wave;  // one head per wave
  const float* q = ws + Q_OFF;
  const float* k = ws + K_OFF;
  float* attn = ws + ATTN_OFF;
  size_t qb = (((size_t)b * Hh + hh) * Nn + l0 + l16) * Dd;
  size_t kb = (((size_t)b * Hh + hh) * Nn + m0 + l16) * Dd;
  v8f c = {};
#pragma unroll
  for (int kk = 0; kk < Dd; kk += 4) {
    int ka = kk + 2 * half;
    v2f a, bv;
    a.x = q[qb + ka];  a.y = q[qb + ka + 1];
    bv.x = k[kb + ka]; bv.y = k[kb + ka + 1];
    c = wmma4(a, bv, c);
  }
#pragma unroll
  for (int j = 0; j < 8; j++) {
    int lr = j + 8 * half;
    float val = fminf(fmaxf(c[j], -5.f), 5.f) + s_ebias[lr][l16][hh];
    attn[((((size_t)b * Hh + hh) * Nn) + l0 + lr) * Nn + m0 + l16] = val;
  }
}

// ---------------- kernel 3: softmax stats + degree scalers ----------------
__global__ __launch_bounds__(256) void k_softmax_stats(float* __restrict__ ws)
{
  int t = threadIdx.x, wave = t >> 5, lane = t & 31;
  int rid = blockIdx.x * 8 + wave;  // (b*H+h)*N + l
  int l = rid & (Nn - 1);
  const float* attn = ws + ATTN_OFF + (size_t)rid * Nn;
  const float* gate = ws + GATE_OFF + (size_t)rid * Nn;
  float mx = -1e30f;
  for (int mi = lane; mi < Nn; mi += 32) mx = fmaxf(mx, attn[mi]);
#pragma unroll
  for (int o = 16; o > 0; o >>= 1) mx = fmaxf(mx, __shfl_xor(mx, o, 32));
  float se = 0.f, sg = 0.f;
  for (int mi = lane; mi < Nn; mi += 32) { se += __expf(attn[mi] - mx); sg += gate[mi]; }
#pragma unroll
  for (int o = 16; o > 0; o >>= 1) { se += __shfl_xor(se, o, 32); sg += __shfl_xor(sg, o, 32); }
  if (lane == 0) {
    ws[RMAX_OFF + rid] = mx;
    ws[RSUM_OFF + rid] = se;
    ws[SCAL_OFF + rid] = (l < NVi) ? 1.f : log1pf(sg);
  }
}

// ---------------- kernel 4: attn_tild @ V ----------------
__global__ __launch_bounds__(128) void k_vattn(float* __restrict__ ws)
{
  __shared__ float acc[4][16][16];
  int bh = blockIdx.x;            // b*H + h
  int l0 = blockIdx.y << 4;
  int t = threadIdx.x, w = t >> 5, lane = t & 31, half = lane >> 4, l16 = lane & 15;
  const float* attn = ws + ATTN_OFF + ((size_t)bh * Nn + l0 + l16) * Nn;
  const float* gate = ws + GATE_OFF + ((size_t)bh * Nn + l0 + l16) * Nn;
  const float* v = ws + V_OFF + (size_t)bh * Nn * Dd;
  float rm = ws[RMAX_OFF + bh * Nn + l0 + l16];
  float ri = 1.f / ws[RSUM_OFF + bh * Nn + l0 + l16];
  v8f c = {};
  for (int mtile = w; mtile < 32; mtile += 4) {
    int m0 = mtile << 4;
#pragma unroll
    for (int kk = 0; kk < 16; kk += 4) {
      int k1 = m0 + kk + 2 * half;
      v2f a, bv;
      a.x = __expf(attn[k1] - rm) * ri * gate[k1];
      a.y = __expf(attn[k1 + 1] - rm) * ri * gate[k1 + 1];
      bv.x = v[(size_t)k1 * Dd + l16];
      bv.y = v[(size_t)(k1 + 1) * Dd + l16];
      c = wmma4(a, bv, c);
    }
  }
#pragma unroll
  for (int j = 0; j < 8; j++) acc[w][j + 8 * half][l16] = c[j];
  __syncthreads();
  int b = bh >> 3, hh = bh & 7;
  float* vatt = ws + VATT_OFF;
  for (int e = t; e < 256; e += 128) {
    int rr = e >> 4, cc = e & 15;
    float sum = acc[0][rr][cc] + acc[1][rr][cc] + acc[2][rr][cc] + acc[3][rr][cc];
    sum *= ws[SCAL_OFF + bh * Nn + l0 + rr];
    vatt[((size_t)b * Nn + l0 + rr) * Ff + cc * Hh + hh] = sum;
  }
}

// ---------------- kernel 5: node projection + FFN ----------------
__global__ __launch_bounds__(256) void k_node_out(
    const float* __restrict__ nfeat, const float* __restrict__ Wno,
    const float* __restrict__ bno, const float* __restrict__ fg,
    const float* __restrict__ fb, const float* __restrict__ Wf1,
    const float* __restrict__ bf1, const float* __restrict__ Wf2,
    const float* __restrict__ bf2, const float* __restrict__ ws,
    float* __restrict__ out)
{
  __shared__ float sA[16][Ff + 1];
  __shared__ float sN[16][Ff + 1];
  __shared__ float sB[16][Ff + 1];
  __shared__ float red1[16][16], red2[16][16], smean[16], srstd[16];
  int blk = blockIdx.x;
  int b = blk >> 5;
  int r0 = (blk & 31) << 4;
  int t = threadIdx.x;
  int row = t >> 4, sub = t & 15;
  const float* vatt = ws + VATT_OFF;
#pragma unroll
  for (int i = 0; i < 8; i++) {
    int col = sub * 8 + i;
    sA[row][col] = vatt[((size_t)b * Nn + r0 + row) * Ff + col];
  }
  __syncthreads();
  int wave = t >> 5, lane = t & 31, half = lane >> 4, l16 = lane & 15;
  int col0 = wave << 4;
  {  // GEMM1: v_attn @ W_no + b_no + nfeat -> sN
    v8f c = {};
#pragma unroll
    for (int kk = 0; kk < Ff; kk += 4) {
      int ka = kk + 2 * half;
      v2f a, bv;
      a.x = sA[l16][ka]; a.y = sA[l16][ka + 1];
      bv.x = Wno[ka * Ff + col0 + l16];
      bv.y = Wno[(ka + 1) * Ff + col0 + l16];
      c = wmma4(a, bv, c);
    }
    int colg = col0 + l16;
#pragma unroll
    for (int j = 0; j < 8; j++) {
      int rr = j + 8 * half;
      sN[rr][colg] = c[j] + bno[colg] + nfeat[((size_t)b * Nn + r0 + rr) * Ff + colg];
    }
  }
  __syncthreads();
  {  // LN(sN) -> sA
    float s = 0.f, s2 = 0.f, vals[8];
#pragma unroll
    for (int i = 0; i < 8; i++) { float x = sN[row][sub * 8 + i]; vals[i] = x; s += x; s2 += x * x; }
    red1[row][sub] = s; red2[row][sub] = s2;
    __syncthreads();
    if (sub == 0) {
      float a = 0.f, c2 = 0.f;
#pragma unroll
      for (int i = 0; i < 16; i++) { a += red1[row][i]; c2 += red2[row][i]; }
      float mm = a / 128.f, vv = c2 / 128.f - mm * mm;
      smean[row] = mm; srstd[row] = rsqrtf(vv + 1e-5f);
    }
    __syncthreads();
    float mm = smean[row], rr = srstd[row];
#pragma unroll
    for (int i = 0; i < 8; i++) {
      int col = sub * 8 + i;
      sA[row][col] = (vals[i] - mm) * rr * fg[col] + fb[col];
    }
  }
  __syncthreads();
  {  // GEMM2 + ELU -> sB
    v8f c = {};
#pragma unroll
    for (int kk = 0; kk < Ff; kk += 4) {
      int ka = kk + 2 * half;
      v2f a, bv;
      a.x = sA[l16][ka]; a.y = sA[l16][ka + 1];
      bv.x = Wf1[ka * Ff + col0 + l16];
      bv.y = Wf1[(ka + 1) * Ff + col0 + l16];
      c = wmma4(a, bv, c);
    }
    int colg = col0 + l16;
#pragma unroll
    for (int j = 0; j < 8; j++) {
      int rr = j + 8 * half;
      float x = c[j] + bf1[colg];
      sB[rr][colg] = x > 0.f ? x : (__expf(x) - 1.f);
    }
  }
  __syncthreads();
  {  // GEMM3 + residual -> out
    v8f c = {};
#pragma unroll
    for (int kk = 0; kk < Ff; kk += 4) {
      int ka = kk + 2 * half;
      v2f a, bv;
      a.x = sB[l16][ka]; a.y = sB[l16][ka + 1];
      bv.x = Wf2[ka * Ff + col0 + l16];
      bv.y = Wf2[(ka + 1) * Ff + col0 + l16];
      c = wmma4(a, bv, c);
    }
    int colg = col0 + l16;
#pragma unroll
    for (int j = 0; j < 8; j++) {
      int rr = j + 8 * half;
      out[((size_t)b * Nn + r0 + rr) * Ff + colg] = c[j] + bf2[colg] + sN[rr][colg];
    }
  }
}

// ---------------- kernel 6: edge projection + edge FFN ----------------
__global__ __launch_bounds__(256) void k_edge_out(
    const float* __restrict__ efeat, const float* __restrict__ Weo,
    const float* __restrict__ beo, const float* __restrict__ eg2,
    const float* __restrict__ eb2, const float* __restrict__ We1,
    const float* __restrict__ be1, const float* __restrict__ We2,
    const float* __restrict__ be2, const float* __restrict__ ws,
    float* __restrict__ out)
{
  __shared__ float sEf[128 * PADW];   // residual slab via TDM (128 entries x 32)
  __shared__ float sE[8][16][Ee + 1];
  __shared__ float sL[8][16][Ee + 1];
  __shared__ float sAc[8][16][Ee + 1];
  int t = threadIdx.x, wave = t >> 5, lane = t & 31, half = lane >> 4, l16 = lane & 15;
  size_t base = (size_t)blockIdx.x * 128;        // flat edge-entry base
  int b = (int)(base / ((size_t)Nn * Nn));
  int l = (int)((base / Nn) % Nn);
  int m0 = (int)(base % Nn) + wave * 16;
  const float* attn = ws + ATTN_OFF;

  // Stream the 16KB contiguous residual slab via the Tensor Data Mover.
  if (t == 0) {
    tdm_load_2d(efeat + base * Ee, sEf, 128 * Ee, 1, 128 * Ee, 1,
                (unsigned long long)(128 * Ee));
  }
  __builtin_amdgcn_s_wait_tensorcnt(0);
  __syncthreads();

  // GEMM: attn_hat(16 entries x H=8) @ W_eo(8x32) + b_eo + efeat -> sE
#pragma unroll
  for (int ct = 0; ct < 2; ++ct) {
    int col0 = ct << 4;
    v8f c = {};
#pragma unroll
    for (int kk = 0; kk < 8; kk += 4) {
      int ka = kk + 2 * half;
      v2f a, bv;
      a.x = attn[((((size_t)b * Hh + ka) * Nn) + l) * Nn + m0 + l16];
      a.y = attn[((((size_t)b * Hh + ka + 1) * Nn) + l) * Nn + m0 + l16];
      bv.x = Weo[ka * Ee + col0 + l16];
      bv.y = Weo[(ka + 1) * Ee + col0 + l16];
      c = wmma4(a, bv, c);
    }
    int colg = col0 + l16;
#pragma unroll
    for (int j = 0; j < 8; j++) {
      int rr = j + 8 * half;
      sE[wave][rr][colg] = c[j] + beo[colg] + sEf[(wave * 16 + rr) * PADW + colg];
    }
  }
  __syncthreads();
  // per-row LN over E=32 (lanes 0-15, one row each)
  if (lane < 16) {
    float s = 0.f, s2 = 0.f;
#pragma unroll
    for (int i = 0; i < 32; i++) { float x = sE[wave][lane][i]; s += x; s2 += x * x; }
    float mm = s / 32.f, vv = s2 / 32.f - mm * mm, rr = rsqrtf(vv + 1e-5f);
#pragma unroll
    for (int i = 0; i < 32; i++)
      sL[wave][lane][i] = (sE[wave][lane][i] - mm) * rr * eg2[i] + eb2[i];
  }
  __syncthreads();
  // GEMM e1 + ELU -> sAc
#pragma unroll
  for (int ct = 0; ct < 2; ++ct) {
    int col0 = ct << 4;
    v8f c = {};
#pragma unroll
    for (int kk = 0; kk < Ee; kk += 4) {
      int ka = kk + 2 * half;
      v2f a, bv;
      a.x = sL[wave][l16][ka]; a.y = sL[wave][l16][ka + 1];
      bv.x = We1[ka * Ee + col0 + l16];
      bv.y = We1[(ka + 1) * Ee + col0 + l16];
      c = wmma4(a, bv, c);
    }
    int colg = col0 + l16;
#pragma unroll
    for (int j = 0; j < 8; j++) {
      int rr = j + 8 * half;
      float x = c[j] + be1[colg];
      sAc[wave][rr][colg] = x > 0.f ? x : (__expf(x) - 1.f);
    }
  }
  __syncthreads();
  // GEMM e2 + residual -> out
#pragma unroll
  for (int ct = 0; ct < 2; ++ct) {
    int col0 = ct << 4;
    v8f c = {};
#pragma unroll
    for (int kk = 0; kk < Ee; kk += 4) {
      int ka = kk + 2 * half;
      v2f a, bv;
      a.x = sAc[wave][l16][ka]; a.y = sAc[wave][l16][ka + 1];
      bv.x = We2[ka * Ee + col0 + l16];
      bv.y = We2[(ka + 1) * Ee + col0 + l16];
      c = wmma4(a, bv, c);
    }
    int colg = col0 + l16;
#pragma unroll
    for (int j = 0; j < 8; j++) {
      int rr = j + 8 * half;
      size_t ent = ((size_t)b * Nn + l) * Nn + m0 + rr;
      out[NODE_SZ + ent * Ee + colg] = c[j] + be2[colg] + sE[wave][rr][colg];
    }
  }
}

extern "C" void kernel_launch(void* const* d_in, const int* in_sizes, int n_in,
                              void* d_out, int out_size, void* d_ws, size_t ws_size,
                              hipStream_t stream)
{
  const float* nfeat    = (const float*)d_in[0];
  const float* efeat    = (const float*)d_in[1];
  const float* ln_h_g   = (const float*)d_in[2];
  const float* ln_h_b   = (const float*)d_in[3];
  const float* ln_e_g   = (const float*)d_in[4];
  const float* ln_e_b   = (const float*)d_in[5];
  const float* W_edge   = (const float*)d_in[6];
  const float* b_edge   = (const float*)d_in[7];
  const float* W_qkv    = (const float*)d_in[8];
  const float* b_qkv    = (const float*)d_in[9];
  const float* W_gate   = (const float*)d_in[10];
  const float* b_gate   = (const float*)d_in[11];
  const float* W_no     = (const float*)d_in[12];
  const float* b_no     = (const float*)d_in[13];
  const float* ffn_ln_g = (const float*)d_in[14];
  const float* ffn_ln_b = (const float*)d_in[15];
  const float* W_f1     = (const float*)d_in[16];
  const float* b_f1     = (const float*)d_in[17];
  const float* W_f2     = (const float*)d_in[18];
  const float* b_f2     = (const float*)d_in[19];
  const float* W_eo     = (const float*)d_in[20];
  const float* b_eo     = (const float*)d_in[21];
  const float* effn_ln_g= (const float*)d_in[22];
  const float* effn_ln_b= (const float*)d_in[23];
  const float* W_e1     = (const float*)d_in[24];
  const float* b_e1     = (const float*)d_in[25];
  const float* W_e2     = (const float*)d_in[26];
  const float* b_e2     = (const float*)d_in[27];
  float* ws  = (float*)d_ws;
  float* out = (float*)d_out;

  hipLaunchKernelGGL(k_node_pre, dim3(256), dim3(256), 0, stream,
                     nfeat, ln_h_g, ln_h_b, W_qkv, b_qkv, ws);
  hipLaunchKernelGGL(k_edge_scores, dim3(8, 32, 32), dim3(256), 0, stream,
                     efeat, ln_e_g, ln_e_b, W_edge, b_edge, W_gate, b_gate, ws);
  hipLaunchKernelGGL(k_softmax_stats, dim3(4096), dim3(256), 0, stream, ws);
  hipLaunchKernelGGL(k_vattn, dim3(64, 32), dim3(128), 0, stream, ws);
  hipLaunchKernelGGL(k_node_out, dim3(256), dim3(256), 0, stream,
                     nfeat, W_no, b_no, ffn_ln_g, ffn_ln_b, W_f1, b_f1, W_f2, b_f2,
                     ws, out);
  hipLaunchKernelGGL(k_edge_out, dim3(16384), dim3(256), 0, stream,
                     efeat, W_eo, b_eo, effn_ln_g, effn_ln_b, W_e1, b_e1, W_e2, b_e2,
                     ws, out);
}